// SetAbstraction_48945447305225
// MI455X (gfx1250) — compile-verified
//
#include <hip/hip_runtime.h>
#include <math.h>

typedef float v2f __attribute__((ext_vector_type(2)));
typedef float v8f __attribute__((ext_vector_type(8)));

#define B_    4
#define N_    8192
#define CIN   128
#define COUT  256
#define M_    2048      // N_/STRIDE
#define K_    32        // NSAMPLE
#define NDIR  32
#define HALF  128
#define R2    0.01f     // RADIUS^2
#define BNEPS 1e-5f

// ---------------------------------------------------------------------------
// 1) Farthest point sampling: one workgroup per batch, points in registers,
//    wave32 shfl argmax + LDS cross-wave reduction per round.
// ---------------------------------------------------------------------------
#define FPS_THREADS 1024
#define PPT (N_ / FPS_THREADS)   // 8 points per thread

__global__ __launch_bounds__(FPS_THREADS)
void fps_kernel(const float* __restrict__ p, int* __restrict__ idx)
{
    const int b    = blockIdx.x;
    const int tid  = threadIdx.x;
    const int lane = tid & 31;
    const int wave = tid >> 5;
    const float* pb = p + (size_t)b * N_ * 3;

    float px[PPT], py[PPT], pz[PPT], dist[PPT];
#pragma unroll
    for (int j = 0; j < PPT; ++j) {
        int n = tid * PPT + j;
        px[j] = pb[n * 3 + 0];
        py[j] = pb[n * 3 + 1];
        pz[j] = pb[n * 3 + 2];
        dist[j] = 1e10f;
    }

    __shared__ float s_d[32];
    __shared__ int   s_i[32];
    __shared__ float s_last[3];

    if (tid == 0) {
        idx[b * M_ + 0] = 0;
        s_last[0] = pb[0]; s_last[1] = pb[1]; s_last[2] = pb[2];
    }
    __syncthreads();

    for (int m = 1; m < M_; ++m) {
        const float lx = s_last[0], ly = s_last[1], lz = s_last[2];
        float bestd = -1.0f; int besti = 0;
#pragma unroll
        for (int j = 0; j < PPT; ++j) {
            float dx = px[j] - lx, dy = py[j] - ly, dz = pz[j] - lz;
            float d  = dx * dx + dy * dy + dz * dz;
            d = fminf(d, dist[j]);
            dist[j] = d;
            if (d > bestd) { bestd = d; besti = tid * PPT + j; }
        }
        // intra-wave argmax (wave32)
#pragma unroll
        for (int off = 16; off > 0; off >>= 1) {
            float od = __shfl_down(bestd, off, 32);
            int   oi = __shfl_down(besti, off, 32);
            if (od > bestd) { bestd = od; besti = oi; }
        }
        if (lane == 0) { s_d[wave] = bestd; s_i[wave] = besti; }
        __syncthreads();
        if (wave == 0) {
            float bd = s_d[lane]; int bi = s_i[lane];
#pragma unroll
            for (int off = 16; off > 0; off >>= 1) {
                float od = __shfl_down(bd, off, 32);
                int   oi = __shfl_down(bi, off, 32);
                if (od > bd) { bd = od; bi = oi; }
            }
            if (lane == 0) {
                idx[b * M_ + m] = bi;
                s_last[0] = pb[bi * 3 + 0];
                s_last[1] = pb[bi * 3 + 1];
                s_last[2] = pb[bi * 3 + 2];
            }
        }
        __syncthreads();
    }
}

// ---------------------------------------------------------------------------
// 2) Gathers
// ---------------------------------------------------------------------------
__global__ void gather_newp_kernel(const float* __restrict__ p,
                                   const int* __restrict__ idx,
                                   float* __restrict__ new_p)
{
    int t = blockIdx.x * blockDim.x + threadIdx.x;
    if (t >= B_ * M_) return;
    int b = t / M_;
    int i = idx[t];
    const float* src = p + ((size_t)b * N_ + i) * 3;
    new_p[t * 3 + 0] = src[0];
    new_p[t * 3 + 1] = src[1];
    new_p[t * 3 + 2] = src[2];
}

__global__ void gather_fi_kernel(const float* __restrict__ f,
                                 const int* __restrict__ idx,
                                 float* __restrict__ fi)   // B x CIN x M
{
    int t = blockIdx.x * blockDim.x + threadIdx.x;
    if (t >= B_ * CIN * M_) return;
    int b = t / (CIN * M_);
    int r = t % (CIN * M_);
    int c = r / M_;
    int m = r % M_;
    fi[t] = f[((size_t)b * CIN + c) * N_ + idx[b * M_ + m]];
}

// ---------------------------------------------------------------------------
// 3) fp32 WMMA GEMM:  Y[b] = W(OxC) @ X[b](CxNCOLS)  (+fused epilogue)
//    Compile-time C/NCOLS/O/MODE -> fully unrolled K loop, immediate offsets.
//    Each wave owns a 16x32 output tile (two 16x16 accumulators) so each
//    A-fragment feeds two V_WMMA_F32_16X16X4_F32 ops. 8 waves/block share the
//    16xC W-tile via LDS (row stride padded to C+4 -> conflict-free).
//    Grid covers exactly: EXEC all-ones at every WMMA (ISA requirement).
// ---------------------------------------------------------------------------
#define GEMM_BIAS     0   // Y[b][o][n]  = acc + bias
#define GEMM_BIAS_T   1   // Y[b][n][o]  = acc + bias   (transposed store)
#define GEMM_BN_GELU  2   // Y[b][o][n]  = gelu(bn(acc))
#define GEMM_BIAS_ACC 3   // Y[b][o][n] += acc + bias

template<int O, int NCOLS, int MODE>
__device__ __forceinline__ void gemm_epilogue(
    float v, int row, int col, int bz,
    const float* __restrict__ bias,
    const float* __restrict__ g,  const float* __restrict__ bb,
    const float* __restrict__ mn, const float* __restrict__ vr,
    float* __restrict__ Y)
{
    if (MODE == GEMM_BIAS) {
        Y[((size_t)bz * O + row) * NCOLS + col] = v + bias[row];
    } else if (MODE == GEMM_BIAS_T) {
        Y[((size_t)bz * NCOLS + col) * O + row] = v + bias[row];
    } else if (MODE == GEMM_BN_GELU) {
        float sc = g[row] * rsqrtf(vr[row] + BNEPS);
        float x  = (v - mn[row]) * sc + bb[row];
        Y[((size_t)bz * O + row) * NCOLS + col] =
            0.5f * x * (1.0f + erff(x * 0.70710678118654752f));
    } else { // GEMM_BIAS_ACC
        Y[((size_t)bz * O + row) * NCOLS + col] += v + bias[row];
    }
}

template<int C, int NCOLS, int O, int MODE>
__global__ __launch_bounds__(256)
void wmma_gemm_kernel(const float* __restrict__ W,      // O x C
                      const float* __restrict__ X,      // B x C x NCOLS
                      const float* __restrict__ bias,   // O (unused by BN_GELU)
                      float* __restrict__ Y,
                      const float* __restrict__ g,  const float* __restrict__ bb,
                      const float* __restrict__ mn, const float* __restrict__ vr)
{
    constexpr int LDW = C + 4;               // padded LDS row stride (floats)
    __shared__ float sW[16 * LDW];
    const int lane = threadIdx.x & 31;
    const int wave = threadIdx.x >> 5;
    const int o0   = blockIdx.y * 16;
    const int n0   = blockIdx.x * 256 + wave * 32;   // 32 columns per wave
    const int bz   = blockIdx.z;

    for (int t = threadIdx.x; t < 16 * C; t += 256)
        sW[(t / C) * LDW + (t % C)] = W[(o0 + t / C) * C + (t % C)];
    __syncthreads();

    // A frag (16x4 f32): lanes 0-15 hold K=kk+0,kk+1 ; lanes 16-31 hold kk+2,kk+3
    const int am    = lane & 15;
    const int khalf = (lane >> 4) << 1;      // 0 or 2

    v8f acc0 = {}, acc1 = {};
    const float* Xb = X + (size_t)bz * C * NCOLS + n0 + am;
#pragma unroll
    for (int kk = 0; kk < C; kk += 4) {
        const int kb = kk + khalf;
        v2f a, b0, b1;
        a[0]  = sW[am * LDW + kb];
        a[1]  = sW[am * LDW + kb + 1];
        b0[0] = Xb[(size_t)kb * NCOLS];
        b0[1] = Xb[(size_t)(kb + 1) * NCOLS];
        b1[0] = Xb[(size_t)kb * NCOLS + 16];
        b1[1] = Xb[(size_t)(kb + 1) * NCOLS + 16];
        acc0 = __builtin_amdgcn_wmma_f32_16x16x4_f32(
                   false, a, false, b0, (short)0, acc0, false, false);
        acc1 = __builtin_amdgcn_wmma_f32_16x16x4_f32(
                   false, a, false, b1, (short)0, acc1, false, false);
    }

    // C/D layout: VGPR r -> rows (o0+r) [lanes 0-15] and (o0+8+r) [lanes 16-31]
    const int rbase = (lane >> 4) << 3;
    const int col0  = n0 + am;
#pragma unroll
    for (int r = 0; r < 8; ++r) {
        const int row = o0 + rbase + r;
        gemm_epilogue<O, NCOLS, MODE>(acc0[r], row, col0,      bz, bias, g, bb, mn, vr, Y);
        gemm_epilogue<O, NCOLS, MODE>(acc1[r], row, col0 + 16, bz, bias, g, bb, mn, vr, Y);
    }
}

// ---------------------------------------------------------------------------
// 4) Ball query: one wave per query; ballot + prefix popcount compacts the
//    first K in-radius indices in ascending order (matches jnp.sort semantics).
// ---------------------------------------------------------------------------
__global__ __launch_bounds__(256)
void ball_query_kernel(const float* __restrict__ p,
                       const float* __restrict__ new_p,
                       int* __restrict__ qidx)     // (B*M) x K
{
    const int lane = threadIdx.x & 31;
    const int wave = threadIdx.x >> 5;
    const int q    = blockIdx.x * 8 + wave;
    if (q >= B_ * M_) return;
    const int b = q / M_;
    const float qx = new_p[q * 3 + 0];
    const float qy = new_p[q * 3 + 1];
    const float qz = new_p[q * 3 + 2];
    const float* pb = p + (size_t)b * N_ * 3;

    int cnt = 0, first = -1;
    for (int base = 0; base < N_ && cnt < K_; base += 32) {
        const int n = base + lane;
        float dx = pb[n * 3 + 0] - qx;
        float dy = pb[n * 3 + 1] - qy;
        float dz = pb[n * 3 + 2] - qz;
        float d2 = dx * dx + dy * dy + dz * dz;
        bool in  = d2 < R2;
        unsigned mask = __builtin_amdgcn_ballot_w32(in);
        if (first < 0 && mask) first = base + __builtin_ctz(mask);
        int pos = cnt + __popc(mask & ((1u << lane) - 1u));
        if (in && pos < K_) qidx[q * K_ + pos] = n;
        cnt += __popc(mask);
    }
    // pad unused slots with the first hit (query point itself is always a hit)
    for (int s = cnt + lane; s < K_; s += 32) qidx[q * K_ + s] = first;
}

// ---------------------------------------------------------------------------
// 5) Directional encoding: lane = direction; broadcast 32 normalized offsets
//    through shfl, running max per direction. tmax layout: B x NDIR x M.
// ---------------------------------------------------------------------------
__global__ __launch_bounds__(256)
void dir_tmax_kernel(const float* __restrict__ p,
                     const float* __restrict__ new_p,
                     const float* __restrict__ dirs,   // NDIR x 3
                     const int* __restrict__ qidx,
                     float* __restrict__ tmax)
{
    const int lane = threadIdx.x & 31;
    const int wave = threadIdx.x >> 5;
    const int q    = blockIdx.x * 8 + wave;
    if (q >= B_ * M_) return;
    const int b = q / M_, m = q % M_;

    float vx = dirs[lane * 3 + 0], vy = dirs[lane * 3 + 1], vz = dirs[lane * 3 + 2];
    float vn = fmaxf(sqrtf(vx * vx + vy * vy + vz * vz), 1e-12f);
    vx /= vn; vy /= vn; vz /= vn;

    const float qx = new_p[q * 3 + 0];
    const float qy = new_p[q * 3 + 1];
    const float qz = new_p[q * 3 + 2];
    const int gi = qidx[q * K_ + lane];
    const float* pb = p + (size_t)b * N_ * 3;
    float dx = pb[gi * 3 + 0] - qx;
    float dy = pb[gi * 3 + 1] - qy;
    float dz = pb[gi * 3 + 2] - qz;
    float dn = fmaxf(sqrtf(dx * dx + dy * dy + dz * dz), 1e-12f);
    dx /= dn; dy /= dn; dz /= dn;

    float tm = -1e30f;
#pragma unroll
    for (int k = 0; k < K_; ++k) {
        float bx = __shfl(dx, k, 32);
        float by = __shfl(dy, k, 32);
        float bz = __shfl(dz, k, 32);
        tm = fmaxf(tm, vx * bx + vy * by + vz * bz);
    }
    tmax[((size_t)b * NDIR + lane) * M_ + m] = tm;
}

// ---------------------------------------------------------------------------
// 6) fj max over K + BN, accumulated into out.  f1 is stored transposed
//    (B x N x COUT) so the gather reads 256 contiguous channels per sample
//    (fully coalesced, L2-resident).
// ---------------------------------------------------------------------------
__global__ __launch_bounds__(COUT)
void fjmax_bn_kernel(const float* __restrict__ f1t,   // B x N x COUT
                     const int* __restrict__ qidx,
                     const float* __restrict__ g,  const float* __restrict__ bb,
                     const float* __restrict__ mn, const float* __restrict__ vr,
                     float* __restrict__ fout)         // B x COUT x M (+=)
{
    const int q = blockIdx.x;          // B*M blocks
    const int o = threadIdx.x;         // channel
    const int b = q / M_, m = q % M_;

    __shared__ int sidx[K_];
    if (o < K_) sidx[o] = qidx[q * K_ + o];
    __syncthreads();

    float v = -1e30f;
#pragma unroll 4
    for (int k = 0; k < K_; ++k)
        v = fmaxf(v, f1t[((size_t)b * N_ + sidx[k]) * COUT + o]);

    float sc  = g[o] * rsqrtf(vr[o] + BNEPS);
    float val = (v - mn[o]) * sc + bb[o];
    fout[((size_t)b * COUT + o) * M_ + m] += val;
}

// ---------------------------------------------------------------------------
// Launcher
// ---------------------------------------------------------------------------
extern "C" void kernel_launch(void* const* d_in, const int* in_sizes, int n_in,
                              void* d_out, int out_size, void* d_ws, size_t ws_size,
                              hipStream_t stream)
{
    const float* p       = (const float*)d_in[0];
    const float* f       = (const float*)d_in[1];
    const float* conv1_w = (const float*)d_in[2];
    const float* conv1_b = (const float*)d_in[3];
    const float* skip_w  = (const float*)d_in[4];
    const float* skip_b  = (const float*)d_in[5];
    const float* bn1_g   = (const float*)d_in[6];
    const float* bn1_b   = (const float*)d_in[7];
    const float* bn1_m   = (const float*)d_in[8];
    const float* bn1_v   = (const float*)d_in[9];
    const float* dirs    = (const float*)d_in[10];
    const float* de_w1   = (const float*)d_in[11];
    const float* de_g    = (const float*)d_in[12];
    const float* de_bb   = (const float*)d_in[13];
    const float* de_m    = (const float*)d_in[14];
    const float* de_v    = (const float*)d_in[15];
    const float* de_w2   = (const float*)d_in[16];
    const float* de_b2   = (const float*)d_in[17];

    float* out_newp = (float*)d_out;                       // B*M*3
    float* out_f    = out_newp + (size_t)B_ * M_ * 3;      // B*COUT*M

    // workspace carve-up (~44 MB total)
    char*  ws    = (char*)d_ws;
    int*   idx   = (int*)ws;    ws += sizeof(int)   * (size_t)B_ * M_;
    int*   qidx  = (int*)ws;    ws += sizeof(int)   * (size_t)B_ * M_ * K_;
    float* fi    = (float*)ws;  ws += sizeof(float) * (size_t)B_ * CIN  * M_;
    float* tmaxb = (float*)ws;  ws += sizeof(float) * (size_t)B_ * NDIR * M_;
    float* hbuf  = (float*)ws;  ws += sizeof(float) * (size_t)B_ * HALF * M_;
    float* f1t   = (float*)ws;  ws += sizeof(float) * (size_t)B_ * N_   * COUT;

    fps_kernel<<<B_, FPS_THREADS, 0, stream>>>(p, idx);
    gather_newp_kernel<<<(B_ * M_ + 255) / 256, 256, 0, stream>>>(p, idx, out_newp);
    gather_fi_kernel<<<(B_ * CIN * M_) / 256, 256, 0, stream>>>(f, idx, fi);

    // identity = skip_w @ fi + skip_b   (overwrites out_f)
    wmma_gemm_kernel<CIN, M_, COUT, GEMM_BIAS>
        <<<dim3(M_ / 256, COUT / 16, B_), 256, 0, stream>>>(
        skip_w, fi, skip_b, out_f, nullptr, nullptr, nullptr, nullptr);

    // f1^T = (conv1_w @ f + conv1_b)^T  -> B x N x COUT
    wmma_gemm_kernel<CIN, N_, COUT, GEMM_BIAS_T>
        <<<dim3(N_ / 256, COUT / 16, B_), 256, 0, stream>>>(
        conv1_w, f, conv1_b, f1t, nullptr, nullptr, nullptr, nullptr);

    ball_query_kernel<<<(B_ * M_) / 8, 256, 0, stream>>>(p, out_newp, qidx);
    dir_tmax_kernel<<<(B_ * M_) / 8, 256, 0, stream>>>(p, out_newp, dirs, qidx, tmaxb);

    // h = gelu(bn(de_w1 @ tmax))
    wmma_gemm_kernel<NDIR, M_, HALF, GEMM_BN_GELU>
        <<<dim3(M_ / 256, HALF / 16, B_), 256, 0, stream>>>(
        de_w1, tmaxb, nullptr, hbuf, de_g, de_bb, de_m, de_v);

    // out_f += de_w2 @ h + de_b2
    wmma_gemm_kernel<HALF, M_, COUT, GEMM_BIAS_ACC>
        <<<dim3(M_ / 256, COUT / 16, B_), 256, 0, stream>>>(
        de_w2, hbuf, de_b2, out_f, nullptr, nullptr, nullptr, nullptr);

    // out_f += bn1(max_k f1[..., qidx])
    fjmax_bn_kernel<<<B_ * M_, COUT, 0, stream>>>(
        f1t, qidx, bn1_g, bn1_b, bn1_m, bn1_v, out_f);
}